// GNNEncoder_40785009442961
// MI455X (gfx1250) — compile-verified
//
#include <hip/hip_runtime.h>
#include <math.h>

#define BATCH   16
#define NPER    400
#define NTOT    (BATCH*NPER)     // 6400
#define KNN     20
#define HEADS   8
#define HDIM    16
#define EMB     128
#define LAYERS  3
#define NEG_SLOPE 0.2f
#define BN_EPS  1e-5f
#define AWORDS  13               // ceil(400/32) bitmask words per adjacency row

typedef __attribute__((ext_vector_type(2)))  float        v2f;
typedef __attribute__((ext_vector_type(8)))  float        v8f;
typedef __attribute__((ext_vector_type(4)))  unsigned int u32x4;
typedef __attribute__((ext_vector_type(4)))  int          i32x4;
typedef __attribute__((ext_vector_type(8)))  int          i32x8;

// ---------------------------------------------------------------- utilities
__global__ void k_zero_u32(unsigned int* p, int n) {
  int i = blockIdx.x * blockDim.x + threadIdx.x;
  if (i < n) p[i] = 0u;
}

// ------------------------------------------------- kNN adjacency (wave32/node)
// One 32-lane wave per node. Lanes own candidates j = lane + 32*t (t<13).
// 20 rounds of wave-wide min-reduction; winner marked consumed; adjacency
// built as OR of directed kNN + transpose + self loop via atomicOr bitmask.
__global__ void k_knn_adj(const float* __restrict__ pos,
                          unsigned int* __restrict__ adj) {
  const int node = blockIdx.x;                 // 0..6399
  const int g = node / NPER, li = node % NPER;
  const int lane = threadIdx.x;                // 0..31 (wave32)
  const float px = pos[node * 2 + 0], py = pos[node * 2 + 1];

  float d[AWORDS];
  int   cand[AWORDS];
#pragma unroll
  for (int t = 0; t < AWORDS; ++t) {
    int jl = lane + 32 * t;
    float dv = 3.0e38f;
    if (jl < NPER && jl != li) {
      int j = g * NPER + jl;
      float dx = px - pos[j * 2 + 0];
      float dy = py - pos[j * 2 + 1];
      dv = dx * dx + dy * dy;
    }
    d[t] = dv; cand[t] = jl;
  }

  unsigned int* rowbase = adj + (size_t)g * NPER * AWORDS;
  for (int k = 0; k < KNN; ++k) {
    float best = 3.0e38f; int bidx = -1;
#pragma unroll
    for (int t = 0; t < AWORDS; ++t)
      if (d[t] < best) { best = d[t]; bidx = cand[t]; }
    // wave32 min-reduce (value, index)
    for (int off = 16; off >= 1; off >>= 1) {
      float ob = __shfl_xor(best, off, 32);
      int   oi = __shfl_xor(bidx, off, 32);
      if (ob < best || (ob == best && oi >= 0 && oi < bidx)) { best = ob; bidx = oi; }
    }
    if (bidx >= 0 && (bidx & 31) == lane) d[bidx >> 5] = 3.0e38f; // consume
    if (lane == 0 && bidx >= 0) {
      atomicOr(&rowbase[(size_t)li * AWORDS + (bidx >> 5)], 1u << (bidx & 31));
      atomicOr(&rowbase[(size_t)bidx * AWORDS + (li >> 5)], 1u << (li & 31));
    }
  }
  if (lane == 0)   // GATConv self loop
    atomicOr(&rowbase[(size_t)li * AWORDS + (li >> 5)], 1u << (li & 31));
}

// -------------------------------------------------------- input projection
__global__ void k_proj(const float* __restrict__ x, const float* __restrict__ w,
                       const float* __restrict__ b, float* __restrict__ h) {
  int i = blockIdx.x * blockDim.x + threadIdx.x;
  if (i >= NTOT * EMB) return;
  int n = i >> 7, c = i & 127;
  float acc = b[c];
#pragma unroll
  for (int k = 0; k < 3; ++k) acc += x[n * 3 + k] * w[k * EMB + c];
  h[i] = acc;
}

// ----------------------------------------------- 6400x128 @ 128x128 via WMMA
// One block (8 waves) per 16-row tile; wave w owns 16-col tile w. The whole
// 64KB fp32 weight matrix is staged into LDS once per block by the Tensor
// Data Mover (one DMA descriptor, synchronized via TENSORcnt), so the inner
// loop reads B from LDS and A as coalesced b64 pairs. fp32 matrix pipe
// (v_wmma_f32_16x16x4_f32) keeps exact reference precision.
__global__ void k_gemm_wmma(const float* __restrict__ A,
                            const float* __restrict__ W,
                            float* __restrict__ C) {
  __shared__ float lds_w[EMB * EMB];           // 64 KB of 320 KB/WGP

  // ---- stage W -> LDS ----
#if defined(__AMDGCN__) && __has_builtin(__builtin_amdgcn_tensor_load_to_lds)
  if ((threadIdx.x >> 5) == 0) {               // one DMA per block (TDM ignores EXEC)
    unsigned lds_off = (unsigned)(uintptr_t)(void*)lds_w;  // LDS addr = low 32 bits
    unsigned long long ga = (unsigned long long)(uintptr_t)W;
    u32x4 g0 = {0u, 0u, 0u, 0u};
    g0[0] = 1u;                                            // count=1, user mode
    g0[1] = lds_off;                                       // D#.lds_addr (bytes)
    g0[2] = (unsigned)(ga & 0xffffffffull);                // global_addr[31:0]
    g0[3] = (unsigned)((ga >> 32) & 0x1ffffffull)          // global_addr[56:32]
            | (2u << 30);                                  // type=2 ("image")
    i32x8 g1 = {0, 0, 0, 0, 0, 0, 0, 0};
    g1[0] = (int)(2u << 16);                   // wg_mask=0, data_size=2 (4B)
    g1[1] = (int)(((unsigned)EMB) << 16);      // tensor_dim0 = 128
    g1[2] = (int)(((unsigned)EMB) << 16);      // tensor_dim1 = 128
    g1[3] = (int)(((unsigned)EMB) << 16);      // tile_dim0   = 128
    g1[4] = (int)EMB;                          // tile_dim1   = 128, tile_dim2=0
    g1[5] = (int)EMB;                          // tensor_dim0_stride = 128
    i32x4 z4 = {0, 0, 0, 0};
#if __clang_major__ >= 23
    i32x8 z8 = {0, 0, 0, 0, 0, 0, 0, 0};
    __builtin_amdgcn_tensor_load_to_lds(g0, g1, z4, z4, z8, 0);
#else
    __builtin_amdgcn_tensor_load_to_lds(g0, g1, z4, z4, 0);
#endif
    __builtin_amdgcn_s_wait_tensorcnt(0);
  }
#else
  for (int idx = threadIdx.x; idx < EMB * EMB; idx += 256) lds_w[idx] = W[idx];
#endif
  __syncthreads();

  const int rowTile = blockIdx.x;              // 0..399
  const int colTile = threadIdx.x >> 5;        // 0..7 (one wave per col tile)
  const int lane    = threadIdx.x & 31;
  const int m       = lane & 15;
  const int khalf   = (lane >> 4) << 1;        // 0 or 2 (K sub-pair)
  const int row     = rowTile * 16 + m;
  const int col     = colTile * 16 + (lane & 15);
  const int mbase   = rowTile * 16 + ((lane >> 4) << 3);

#if defined(__AMDGCN__) && __has_builtin(__builtin_amdgcn_wmma_f32_16x16x4_f32)
  v8f c = {0.f, 0.f, 0.f, 0.f, 0.f, 0.f, 0.f, 0.f};
  for (int k0 = 0; k0 < EMB; k0 += 4) {
    v2f a = *(const v2f*)(A + (size_t)row * EMB + k0 + khalf);  // coalesced pair
    v2f b;
    b.x = lds_w[(k0 + khalf + 0) * EMB + col];                  // LDS broadcast
    b.y = lds_w[(k0 + khalf + 1) * EMB + col];
    c = __builtin_amdgcn_wmma_f32_16x16x4_f32(false, a, false, b,
                                              (short)0, c, false, false);
  }
#pragma unroll
  for (int r = 0; r < 8; ++r)
    C[(size_t)(mbase + r) * EMB + col] = c[r];
#else
  // host-pass / fallback body (never codegen'd for gfx1250)
  for (int r = 0; r < 8; ++r) {
    float acc = 0.f;
    for (int k = 0; k < EMB; ++k)
      acc += A[(size_t)(mbase + r) * EMB + k] * lds_w[k * EMB + col];
    C[(size_t)(mbase + r) * EMB + col] = acc;
  }
#endif
}

// ------------------------------------------- per-(node,head) attention scores
__global__ void k_scores(const float* __restrict__ hh,
                         const float* __restrict__ a_src,
                         const float* __restrict__ a_dst,
                         float* __restrict__ s_src, float* __restrict__ s_dst) {
  int i = blockIdx.x * blockDim.x + threadIdx.x;
  if (i >= NTOT * HEADS) return;
  int n = i >> 3, hd = i & 7;
  const float* v = hh + (size_t)n * EMB + hd * HDIM;
  float ss = 0.f, sd = 0.f;
#pragma unroll
  for (int d0 = 0; d0 < HDIM; ++d0) {
    ss += v[d0] * a_src[hd * HDIM + d0];
    sd += v[d0] * a_dst[hd * HDIM + d0];
  }
  s_src[i] = ss; s_dst[i] = sd;
}

// ------------------- sparse masked-softmax aggregation (online softmax, ctz)
// One block of 128 threads per destination node i; thread t owns output dim t
// (head = t>>4). All threads scan the same ~41 neighbor bits in lock-step.
__global__ void k_attn_agg(const unsigned int* __restrict__ adj,
                           const float* __restrict__ hh,
                           const float* __restrict__ s_src,
                           const float* __restrict__ s_dst,
                           const float* __restrict__ bias,
                           float* __restrict__ out) {
  __shared__ unsigned int arow[AWORDS];
  const int i = blockIdx.x;
  const int g = i / NPER, li = i % NPER;
  const int t = threadIdx.x;                   // 0..127
  if (t < AWORDS) arow[t] = adj[((size_t)g * NPER + li) * AWORDS + t];
  __syncthreads();
  const int hd = t >> 4;
  const float sdi = s_dst[i * HEADS + hd];
  float mx = -3.0e38f, ssum = 0.f, acc = 0.f;
  for (int w = 0; w < AWORDS; ++w) {
    unsigned int bits = arow[w];
    while (bits) {
      int jl = (w << 5) + __builtin_ctz(bits);
      bits &= bits - 1u;
      int j = g * NPER + jl;
      float sc = sdi + s_src[j * HEADS + hd];
      sc = sc > 0.f ? sc : NEG_SLOPE * sc;     // LeakyReLU(0.2)
      float nm = sc > mx ? sc : mx;
      float e0 = __expf(mx - nm);              // 0 on first neighbor
      float e1 = __expf(sc - nm);
      ssum = ssum * e0 + e1;
      acc  = acc  * e0 + e1 * hh[(size_t)j * EMB + t];
      mx = nm;
    }
  }
  out[(size_t)i * EMB + t] = acc / ssum + bias[t];
}

// ----------------------------------------------------- BatchNorm statistics
__global__ void k_bn_stats(const float* __restrict__ x,
                           float* __restrict__ mean, float* __restrict__ var) {
  __shared__ float s1[256], s2[256];
  const int c = blockIdx.x;
  float a = 0.f, b = 0.f;
  for (int n = threadIdx.x; n < NTOT; n += 256) {
    float v = x[(size_t)n * EMB + c];
    a += v; b += v * v;
  }
  s1[threadIdx.x] = a; s2[threadIdx.x] = b;
  __syncthreads();
  for (int off = 128; off >= 1; off >>= 1) {
    if (threadIdx.x < off) {
      s1[threadIdx.x] += s1[threadIdx.x + off];
      s2[threadIdx.x] += s2[threadIdx.x + off];
    }
    __syncthreads();
  }
  if (threadIdx.x == 0) {
    float mu = s1[0] * (1.f / NTOT);
    mean[c] = mu;
    var[c]  = s2[0] * (1.f / NTOT) - mu * mu;   // biased variance
  }
}

// ------------------------------------------- BN-normalize + ELU + residual
__global__ void k_bn_elu_res(const float* __restrict__ gout,
                             const float* __restrict__ mean,
                             const float* __restrict__ var,
                             const float* __restrict__ gamma,
                             const float* __restrict__ beta,
                             float* __restrict__ h) {
  int i = blockIdx.x * blockDim.x + threadIdx.x;
  if (i >= NTOT * EMB) return;
  int c = i & 127;
  float xn = (gout[i] - mean[c]) * rsqrtf(var[c] + BN_EPS) * gamma[c] + beta[c];
  float e = xn > 0.f ? xn : (__expf(xn) - 1.f);  // ELU
  h[i] = e + h[i];                               // residual (element-local RMW)
}

// ------------------------------------------------------------- finalization
__global__ void k_copy(const float* __restrict__ h, float* __restrict__ out) {
  int i = blockIdx.x * blockDim.x + threadIdx.x;
  if (i < NTOT * EMB) out[i] = h[i];
}

__global__ void k_graph_mean(const float* __restrict__ h, float* __restrict__ out) {
  const int g = blockIdx.x, c = threadIdx.x;     // 16 blocks x 128 threads
  float s = 0.f;
  for (int n = 0; n < NPER; ++n) s += h[(size_t)(g * NPER + n) * EMB + c];
  out[NTOT * EMB + g * EMB + c] = s * (1.0f / NPER);
}

// -------------------------------------------------------------------- launch
extern "C" void kernel_launch(void* const* d_in, const int* in_sizes, int n_in,
                              void* d_out, int out_size, void* d_ws, size_t ws_size,
                              hipStream_t stream) {
  (void)in_sizes; (void)n_in; (void)out_size; (void)ws_size;
  const float* x       = (const float*)d_in[0];
  const float* pos     = (const float*)d_in[1];
  // d_in[2] = batch ids (layout known: 400 nodes per graph, sorted)
  const float* proj_w  = (const float*)d_in[3];
  const float* proj_b  = (const float*)d_in[4];
  const float* lin_w   = (const float*)d_in[5];
  const float* att_src = (const float*)d_in[6];
  const float* att_dst = (const float*)d_in[7];
  const float* gbias   = (const float*)d_in[8];
  const float* gamma   = (const float*)d_in[9];
  const float* beta    = (const float*)d_in[10];
  float* out = (float*)d_out;

  // workspace carve-up (256B-aligned sections, ~10.6 MB total)
  char* ws = (char*)d_ws;
  unsigned int* adj = (unsigned int*)(ws);                       //   332,800 B
  float* h    = (float*)(ws + 332800);                           // 3,276,800 B
  float* hh   = (float*)(ws + 332800 + 1 * 3276800);             // 3,276,800 B
  float* gout = (float*)(ws + 332800 + 2 * 3276800);             // 3,276,800 B
  float* ssrc = (float*)(ws + 332800 + 3 * 3276800);             //   204,800 B
  float* sdst = (float*)(ws + 332800 + 3 * 3276800 + 204800);    //   204,800 B
  float* bnm  = (float*)(ws + 332800 + 3 * 3276800 + 2 * 204800);
  float* bnv  = bnm + 128;

  const int nadj = BATCH * NPER * AWORDS;
  k_zero_u32<<<(nadj + 255) / 256, 256, 0, stream>>>(adj, nadj);
  k_knn_adj<<<NTOT, 32, 0, stream>>>(pos, adj);
  k_proj<<<(NTOT * EMB + 255) / 256, 256, 0, stream>>>(x, proj_w, proj_b, h);

  for (int l = 0; l < LAYERS; ++l) {
    // one block (8 waves, 64KB LDS-staged W via TDM) per 16-row tile
    k_gemm_wmma<<<400, 256, 0, stream>>>(h, lin_w + (size_t)l * EMB * EMB, hh);
    k_scores<<<(NTOT * HEADS + 255) / 256, 256, 0, stream>>>(
        hh, att_src + l * HEADS * HDIM, att_dst + l * HEADS * HDIM, ssrc, sdst);
    k_attn_agg<<<NTOT, 128, 0, stream>>>(adj, hh, ssrc, sdst, gbias + l * EMB, gout);
    k_bn_stats<<<EMB, 256, 0, stream>>>(gout, bnm, bnv);
    k_bn_elu_res<<<(NTOT * EMB + 255) / 256, 256, 0, stream>>>(
        gout, bnm, bnv, gamma + l * EMB, beta + l * EMB, h);
  }

  k_copy<<<(NTOT * EMB + 255) / 256, 256, 0, stream>>>(h, out);
  k_graph_mean<<<BATCH, EMB, 0, stream>>>(h, out);
}